// QuantumGraphConvLayer_30477087932737
// MI455X (gfx1250) — compile-verified
//
#include <hip/hip_runtime.h>
#include <hip/hip_bf16.h>

typedef __attribute__((ext_vector_type(2))) float v2f;
typedef __attribute__((ext_vector_type(8))) float v8f;

// ds_swizzle_b32 with immediate group-of-32 pattern (CDNA5 ISA §11):
//   offset = (xor_mask << 10) | (or_mask << 5) | and_mask, offset[15]=0
//   SWAPX16 -> 0x401F, SWAPX8 -> 0x201F
template <int IMM>
__device__ __forceinline__ float lane_swizzle(float x) {
    int xi = __builtin_bit_cast(int, x);
    int r  = __builtin_amdgcn_ds_swizzle(xi, IMM);
    return __builtin_bit_cast(float, r);
}

// ---------------------------------------------------------------------------
// Kernel 1: build the three 8x8 complex Kronecker matrices
//   M_i = U_{3i} (x) U_{3i+1} (x) U_{3i+2},  U_q = RZ(w2) RY(w1) RX(w0)
// Stored as CM[i][n][k], n=0..15 : n<8 -> Re(M_i[n][k]), n>=8 -> Im(M_i[n-8][k])
// (this is exactly the 16-column WMMA B layout: re in cols 0-7, im in 8-15)
// ---------------------------------------------------------------------------
__global__ void qgc_build_matrices(const float* __restrict__ w,
                                   float* __restrict__ cm) {
    int t = threadIdx.x;
    if (t >= 384) return;
    int i   = t / 128;        // block 0..2
    int rem = t % 128;
    int n   = rem / 8;        // 0..15 (re rows then im rows)
    int k   = rem % 8;        // column
    int row = n & 7;
    // complex product of the three 2x2 entries (qubit 3i is MSB of row/col)
    float re = 1.f, im = 0.f;
    for (int s = 0; s < 3; ++s) {
        int q = 3 * i + s;
        int shift = 2 - s;
        int a = (row >> shift) & 1;   // output bit
        int b = (k   >> shift) & 1;   // input bit
        float hx = 0.5f * w[q * 3 + 0];
        float hy = 0.5f * w[q * 3 + 1];
        float hz = 0.5f * w[q * 3 + 2];
        float cx = cosf(hx), sx = sinf(hx);
        float cy = cosf(hy), sy = sinf(hy);
        // (RY @ RX)[a][b]
        float cre, cim;
        if (a == 0 && b == 0)      { cre =  cy * cx; cim =  sy * sx; }
        else if (a == 0)           { cre = -sy * cx; cim = -cy * sx; }
        else if (b == 0)           { cre =  sy * cx; cim = -cy * sx; }
        else                       { cre =  cy * cx; cim = -sy * sx; }
        // multiply by e^{-i hz} (row 0) or e^{+i hz} (row 1)
        float ez = (a == 0) ? -hz : hz;
        float ce = cosf(ez), se = sinf(ez);
        float ure = cre * ce - cim * se;
        float uim = cre * se + cim * ce;
        float nre = re * ure - im * uim;
        float nim = re * uim + im * ure;
        re = nre; im = nim;
    }
    cm[(i * 16 + n) * 8 + k] = (n >= 8) ? im : re;
}

// ---------------------------------------------------------------------------
// Kernel 2: per wave of 32 lanes handle 16 nodes.
//   - normalize the three 8-vectors (xor-16 swizzle for the K-half reduction)
//   - a_i = M_i s_i via V_WMMA_F32_16X16X4_F32 (K=8 as 2 chained K=4 steps);
//     B packs Re rows in cols 0-7, Im rows in cols 8-15
//   - p_i = re^2 + im^2 via xor-8 swizzle; lanes c and c+8 hold the identical
//     value, so both store unconditionally to the same LDS address (no EXEC
//     branching)
//   - out[n][j] = p0[g>>6] * p1[(g>>3)&7] * p2[g&7],  g = j ^ (j>>1)
//     written as coalesced float4 stores (global_store_b128).
// Physical-K permutation used consistently for A and B operands:
//   k_phys = hi*4 + step*2 + j   (hi = lane>>4, j = operand VGPR index)
// ---------------------------------------------------------------------------
#define WPB 2  // waves per block

__launch_bounds__(64)
__global__ void qgc_main(const float* __restrict__ nf,
                         const float* __restrict__ cm,
                         float* __restrict__ out, int N) {
    __shared__ float p_lds[WPB][16][24];

    const int lane  = threadIdx.x & 31;
    const int wave  = threadIdx.x >> 5;
    const int gwave = blockIdx.x * WPB + wave;
    const long base = (long)gwave * 16;
    const bool active = (base < (long)N);

    const int m  = lane & 15;   // A row (node in tile) == B column n
    const int hi = lane >> 4;

    if (active) {
        const long node = base + m;

        // ---- B operands: CM[i][n][hi*4 + s*2 + {0,1}] ----------------------
        v2f Bop[3][2];
        #pragma unroll
        for (int i = 0; i < 3; ++i) {
            #pragma unroll
            for (int s = 0; s < 2; ++s) {
                const float* p = cm + (i * 16 + m) * 8 + hi * 4 + s * 2;
                v2f b = { p[0], p[1] };
                Bop[i][s] = b;
            }
        }

        // ---- features: lane grabs k = hi*4 .. hi*4+3 of each 8-vector ------
        float4 f[3];
        if (node < (long)N) {
            const float* p = nf + node * 24 + hi * 4;
            f[0] = *(const float4*)(p);
            f[1] = *(const float4*)(p + 8);
            f[2] = *(const float4*)(p + 16);
        } else {
            f[0] = f[1] = f[2] = make_float4(0.f, 0.f, 0.f, 0.f);
        }

        // ---- per block: normalize, 2x WMMA (K=8), p = re^2 + im^2 ----------
        #pragma unroll
        for (int i = 0; i < 3; ++i) {
            float ssq = f[i].x * f[i].x + f[i].y * f[i].y +
                        f[i].z * f[i].z + f[i].w * f[i].w;
            float tot = ssq + lane_swizzle<0x401F>(ssq);   // SWAPX16: other K-half
            float rn  = rsqrtf(tot);

            v2f A0 = { f[i].x * rn, f[i].y * rn };         // k_phys hi*4+0,1
            v2f A1 = { f[i].z * rn, f[i].w * rn };         // k_phys hi*4+2,3

            v8f acc = {};
            acc = __builtin_amdgcn_wmma_f32_16x16x4_f32(
                      false, A0, false, Bop[i][0], (short)0, acc, false, false);
            acc = __builtin_amdgcn_wmma_f32_16x16x4_f32(
                      false, A1, false, Bop[i][1], (short)0, acc, false, false);

            // D[r + 8*hi][n] lives in acc[r] of lane n+16*hi.
            // cols 0-7 = Re(a_i), cols 8-15 = Im(a_i)
            #pragma unroll
            for (int r = 0; r < 8; ++r) {
                float d  = acc[r];
                float sq = d * d;
                float pv = sq + lane_swizzle<0x201F>(sq);  // SWAPX8: re^2 + im^2
                // lanes c and c+8 hold identical pv -> duplicate same-value
                // store to the same address (branch-free)
                p_lds[wave][r + 8 * hi][i * 8 + (lane & 7)] = pv;
            }
        }
    }

    __syncthreads();   // order LDS writes vs reads (all waves participate)

    if (active) {
        // ---- output: 512 probs/node, Gray-code indexed, float4 stores ------
        #pragma unroll 1
        for (int mm = 0; mm < 16; ++mm) {
            const long nnode = base + mm;
            if (nnode >= (long)N) break;
            const float* pp = &p_lds[wave][mm][0];
            float* o = out + nnode * 512;
            #pragma unroll
            for (int it = 0; it < 4; ++it) {
                int j  = (it * 32 + lane) * 4;
                int g0 = j ^ (j >> 1);
                // within an aligned quad only g's low 2 bits vary
                float a  = pp[g0 >> 6];
                float b  = pp[8 + ((g0 >> 3) & 7)];
                float ab = a * b;
                int g1 = (j + 1) ^ ((j + 1) >> 1);
                int g2 = (j + 2) ^ ((j + 2) >> 1);
                int g3 = (j + 3) ^ ((j + 3) >> 1);
                float4 r;
                r.x = ab * pp[16 + (g0 & 7)];
                r.y = ab * pp[16 + (g1 & 7)];
                r.z = ab * pp[16 + (g2 & 7)];
                r.w = ab * pp[16 + (g3 & 7)];
                *(float4*)(o + j) = r;
            }
        }
    }
}

// ---------------------------------------------------------------------------
extern "C" void kernel_launch(void* const* d_in, const int* in_sizes, int n_in,
                              void* d_out, int out_size, void* d_ws, size_t ws_size,
                              hipStream_t stream) {
    const float* nf = (const float*)d_in[0];   // (N, 24) fp32
    const float* w  = (const float*)d_in[1];   // (9, 3) fp32
    float* out = (float*)d_out;                // (N, 512) fp32
    float* cm  = (float*)d_ws;                 // 384 floats scratch

    int N = in_sizes[0] / 24;

    qgc_build_matrices<<<1, 384, 0, stream>>>(w, cm);

    int nwaves = (N + 15) / 16;
    int blocks = (nwaves + WPB - 1) / WPB;
    qgc_main<<<blocks, 64, 0, stream>>>(nf, cm, out, N);
}